// SurvLoss_26448408609261
// MI455X (gfx1250) — compile-verified
//
#include <hip/hip_runtime.h>
#include <cstdint>

// ---------------- problem constants ----------------
#define N_ELEMS   16777216
#define NSEG      1024
#define BLOCKS_A  1024
#define THREADS_A 256
#define CHUNK     (N_ELEMS / BLOCKS_A)   // 16384 elements per block
#define TILE      2048                   // elements per double-buffered tile
#define NTILES    (CHUNK / TILE)         // 8
#define EPT       (TILE / THREADS_A)     // 8 floats per thread per tile
#define NT_TOTAL  (BLOCKS_A * NTILES)    // 8192 tile sums

// ---------------- CDNA5 async global->LDS path ----------------
#if defined(__AMDGCN__) && __has_builtin(__builtin_amdgcn_global_load_async_to_lds_b128)
#define USE_ASYNC_LDS 1
#else
#define USE_ASYNC_LDS 0
#endif

typedef int v4i_t __attribute__((vector_size(4 * sizeof(int))));
#if USE_ASYNC_LDS
typedef __attribute__((address_space(1))) v4i_t* g_v4i_ptr;
typedef __attribute__((address_space(3))) v4i_t* l_v4i_ptr;
#endif

__device__ __forceinline__ void async_copy_32B(const float* g, float* l) {
#if USE_ASYNC_LDS
  // offset immediate applies to BOTH global and LDS addresses (CDNA5 ISA)
  __builtin_amdgcn_global_load_async_to_lds_b128(
      (g_v4i_ptr)(uintptr_t)g, (l_v4i_ptr)l, 0, 0);
  __builtin_amdgcn_global_load_async_to_lds_b128(
      (g_v4i_ptr)(uintptr_t)g, (l_v4i_ptr)l, 16, 0);
#else
  ((float4*)l)[0] = ((const float4*)g)[0];
  ((float4*)l)[1] = ((const float4*)g)[1];
#endif
}

template <int CNT>
__device__ __forceinline__ void waitAsync() {
#if defined(__AMDGCN__)
#if __has_builtin(__builtin_amdgcn_s_wait_asynccnt)
  __builtin_amdgcn_s_wait_asynccnt(CNT);
#else
  asm volatile("s_wait_asynccnt %0" ::"i"(CNT) : "memory");
#endif
#endif
}

// ---------------- wave32 / block reductions ----------------
__device__ __forceinline__ float waveReduceSum(float v) {
#pragma unroll
  for (int off = 16; off > 0; off >>= 1) v += __shfl_down(v, off, 32);
  return v;
}

template <int NWAVES>
__device__ __forceinline__ float blockReduceSum(float v, float* scratch, int tid) {
  v = waveReduceSum(v);
  const int wid = tid >> 5, lane = tid & 31;
  if (lane == 0) scratch[wid] = v;
  __syncthreads();
  v = (tid < NWAVES) ? scratch[tid] : 0.0f;
  if (wid == 0) v = waveReduceSum(v);
  __syncthreads();
  return v;  // valid in thread 0
}

// ---------------- Pass A: stream 192MB, build per-block tables ----------------
__global__ __launch_bounds__(THREADS_A) void passA(
    const float* __restrict__ outs, const int* __restrict__ te,
    const int* __restrict__ tt, float* __restrict__ tile_sum,
    float* __restrict__ blk_s1, float* __restrict__ blk_obs,
    int* __restrict__ blk_last, int* __restrict__ blk_cnt) {
  __shared__ float s_outs[2][TILE];
  __shared__ int   s_lastT[NSEG];
  __shared__ int   s_cntT[NSEG];
  __shared__ float s_red[32];

  const int tid  = threadIdx.x;
  const int b    = blockIdx.x;
  const int base = b * CHUNK;

  for (int k = tid; k < NSEG; k += THREADS_A) { s_lastT[k] = -1; s_cntT[k] = 0; }

  // prologue: async-load tile 0 of outs into LDS buffer 0 (own 32B slot)
  async_copy_32B(outs + base + tid * EPT, &s_outs[0][tid * EPT]);
  __syncthreads();  // covers table init

  float s1 = 0.0f, obs = 0.0f;

  for (int t = 0; t < NTILES; ++t) {
    const int buf = t & 1;
    if (t + 1 < NTILES) {
      async_copy_32B(outs + base + (t + 1) * TILE + tid * EPT,
                     &s_outs[buf ^ 1][tid * EPT]);
      waitAsync<2>();   // in-order: previous tile's 2 loads complete
    } else {
      waitAsync<0>();
    }

    const int idx0 = base + t * TILE + tid * EPT;
    const int4 e0 = *(const int4*)(te + idx0);
    const int4 e1 = *(const int4*)(te + idx0 + 4);
    const int4 g0 = *(const int4*)(tt + idx0);
    const int4 g1 = *(const int4*)(tt + idx0 + 4);
    const float4 y0 = *(const float4*)&s_outs[buf][tid * EPT];
    const float4 y1 = *(const float4*)&s_outs[buf][tid * EPT + 4];

    float esum = 0.0f;
#define PROC(Y, EV, SG, IDX)                                         \
  do {                                                               \
    esum += __expf(Y);                                               \
    int _s = (SG); _s = ((_s < 0) ? -_s : _s) & (NSEG - 1);          \
    atomicMax(&s_lastT[_s], (IDX));                                  \
    if ((EV) > 0) { atomicAdd(&s_cntT[_s], 1); s1 += (Y); obs += 1.0f; } \
  } while (0)
    PROC(y0.x, e0.x, g0.x, idx0 + 0);
    PROC(y0.y, e0.y, g0.y, idx0 + 1);
    PROC(y0.z, e0.z, g0.z, idx0 + 2);
    PROC(y0.w, e0.w, g0.w, idx0 + 3);
    PROC(y1.x, e1.x, g1.x, idx0 + 4);
    PROC(y1.y, e1.y, g1.y, idx0 + 5);
    PROC(y1.z, e1.z, g1.z, idx0 + 6);
    PROC(y1.w, e1.w, g1.w, idx0 + 7);
#undef PROC

    const float tsum = blockReduceSum<8>(esum, s_red, tid);
    if (tid == 0) tile_sum[b * NTILES + t] = tsum;
  }

  __syncthreads();
  for (int k = tid; k < NSEG; k += THREADS_A) {
    blk_last[b * NSEG + k] = s_lastT[k];
    blk_cnt [b * NSEG + k] = s_cntT[k];
  }
  const float r1 = blockReduceSum<8>(s1, s_red, tid);
  const float r2 = blockReduceSum<8>(obs, s_red, tid);
  if (tid == 0) { blk_s1[b] = r1; blk_obs[b] = r2; }
}

// ---------------- Pass B: scan 8192 tile sums, reduce scalars ----------------
__global__ __launch_bounds__(1024) void passB(
    const float* __restrict__ tile_sum, float* __restrict__ tile_pfx,
    const float* __restrict__ blk_s1, const float* __restrict__ blk_obs,
    float* __restrict__ scalars) {
  __shared__ float sc[1024];
  __shared__ float red[32];
  const int tid = threadIdx.x;

  float v[8];
  float tot = 0.0f;
#pragma unroll
  for (int j = 0; j < 8; ++j) { v[j] = tile_sum[tid * 8 + j]; tot += v[j]; }
  sc[tid] = tot;
  __syncthreads();
  for (int off = 1; off < 1024; off <<= 1) {  // Hillis-Steele inclusive scan
    const float x = (tid >= off) ? sc[tid - off] : 0.0f;
    __syncthreads();
    sc[tid] += x;
    __syncthreads();
  }
  float run = sc[tid] - tot;  // exclusive prefix of thread totals
#pragma unroll
  for (int j = 0; j < 8; ++j) { tile_pfx[tid * 8 + j] = run; run += v[j]; }

  const float a = blk_s1[tid];
  const float o = blk_obs[tid];
  __syncthreads();
  const float ra = blockReduceSum<32>(a, red, tid);
  const float ro = blockReduceSum<32>(o, red, tid);
  if (tid == 0) { scalars[0] = ra; scalars[1] = ro; }
}

// ---------------- Pass C: per-segment last index, count, term ----------------
__global__ __launch_bounds__(256) void passC(
    const float* __restrict__ outs, const float* __restrict__ tile_pfx,
    const int* __restrict__ blk_last, const int* __restrict__ blk_cnt,
    float* __restrict__ seg_term) {
  __shared__ int   sL[256], sC[256];
  __shared__ int   segL[16], segC[16];
  __shared__ float red[8];
  const int tid = threadIdx.x;
  const int s0  = blockIdx.x * 16;
  const int tx = tid & 15, ty = tid >> 4;

  int L = -1, cnt = 0;
  for (int b = ty; b < BLOCKS_A; b += 16) {
    const int off = b * NSEG + s0 + tx;  // 16 consecutive ints per row: coalesced
    L = max(L, blk_last[off]);
    cnt += blk_cnt[off];
  }
  sL[tid] = L; sC[tid] = cnt;
  __syncthreads();
  if (tid < 16) {
    int l = sL[tid], c = sC[tid];
#pragma unroll
    for (int k = 1; k < 16; ++k) { l = max(l, sL[tid + k * 16]); c += sC[tid + k * 16]; }
    segL[tid] = l; segC[tid] = c;
  }
  __syncthreads();

  for (int j = 0; j < 16; ++j) {  // uniform branch across block
    const int Ls = segL[j];
    if (Ls >= 0) {
      const int tileIdx = Ls >> 11;
      const int tbase   = tileIdx << 11;
      float part = 0.0f;
      for (int i = tbase + tid; i <= Ls; i += 256) {
        __builtin_prefetch(outs + i + 256, 0, 1);  // global_prefetch_b8
        part += __expf(outs[i]);
      }
      const float tot = blockReduceSum<8>(part, red, tid);
      if (tid == 0) {
        const float S = tile_pfx[tileIdx] + tot;  // inclusive prefix up to Ls
        seg_term[s0 + j] = fmaxf(logf(S), 0.0f) * (float)segC[j];
      }
    } else if (tid == 0) {
      seg_term[s0 + j] = 0.0f;
    }
    __syncthreads();
  }
}

// ---------------- Pass D: final scalar ----------------
__global__ __launch_bounds__(1024) void passD(const float* __restrict__ seg_term,
                                              const float* __restrict__ scalars,
                                              float* __restrict__ out) {
  __shared__ float red[32];
  const int tid = threadIdx.x;
  const float s2 = blockReduceSum<32>(seg_term[tid], red, tid);
  if (tid == 0) out[0] = (s2 - scalars[0]) / scalars[1];
}

// ---------------- launch ----------------
extern "C" void kernel_launch(void* const* d_in, const int* in_sizes, int n_in,
                              void* d_out, int out_size, void* d_ws, size_t ws_size,
                              hipStream_t stream) {
  (void)in_sizes; (void)n_in; (void)out_size; (void)ws_size;
  const float* outs = (const float*)d_in[0];
  const int*   te   = (const int*)d_in[1];
  const int*   tt   = (const int*)d_in[2];
  float*       out  = (float*)d_out;

  float* ws       = (float*)d_ws;
  float* tile_sum = ws;                        // NT_TOTAL
  float* tile_pfx = tile_sum + NT_TOTAL;       // NT_TOTAL
  float* blk_s1   = tile_pfx + NT_TOTAL;       // BLOCKS_A
  float* blk_obs  = blk_s1 + BLOCKS_A;         // BLOCKS_A
  float* scalars  = blk_obs + BLOCKS_A;        // 8 (pad)
  float* seg_term = scalars + 8;               // NSEG
  int*   blk_last = (int*)(seg_term + NSEG);   // BLOCKS_A*NSEG
  int*   blk_cnt  = blk_last + (size_t)BLOCKS_A * NSEG;

  passA<<<BLOCKS_A, THREADS_A, 0, stream>>>(outs, te, tt, tile_sum, blk_s1,
                                            blk_obs, blk_last, blk_cnt);
  passB<<<1, 1024, 0, stream>>>(tile_sum, tile_pfx, blk_s1, blk_obs, scalars);
  passC<<<NSEG / 16, 256, 0, stream>>>(outs, tile_pfx, blk_last, blk_cnt, seg_term);
  passD<<<1, 1024, 0, stream>>>(seg_term, scalars, out);
}